// SupPixPool_25366076850473
// MI455X (gfx1250) — compile-verified
//
#include <hip/hip_runtime.h>
#include <stdint.h>

// ---------------------------------------------------------------------------
// Superpixel max-pool for MI455X (gfx1250, wave32).
//   img [B,C,H,W] fp32, spx [B,H,W] int32 in [0,K) -> out [B,C,K] fp32
// Memory-bound: 256MB read-once img (NT-hinted b128 stream), 2MB L2-resident
// labels re-read by 128 channel planes. One workgroup per (b,c) plane;
// LDS-privatized ds_max_u32 reduction via a monotonic float<->uint key map;
// explicit depth-2 software pipeline (4 b128 loads in flight while older
// tiles feed the DS pipe); output drained LDS->global with the CDNA5 async
// data mover (global_store_async_from_lds_b128 / s_wait_asynccnt).
// Roofline: ~260MB / 23.3 TB/s ~= 11 us; DS-atomic budget ~3 us << floor.
// ---------------------------------------------------------------------------

typedef unsigned int u32;
typedef u32 u32x4 __attribute__((ext_vector_type(4)));
typedef int i32x4 __attribute__((ext_vector_type(4)));

#define B_SZ 8
#define C_SZ 128
#define H_SZ 256
#define W_SZ 256
#define HW_SZ (H_SZ * W_SZ)             // 65536 pixels per plane
#define K_SZ 1024
#define NTHREADS 256                     // 8 wave32s per workgroup
#define ITERS ((HW_SZ / 4) / NTHREADS)   // 64 tiles of 256 x uint4

// Monotonic map: float total order == unsigned total order on keys.
//   key(+x) = bits ^ 0x80000000 ; key(-x) = ~bits
// key(-inf) = 0x007FFFFF (our init value == segment_max identity).
// (Compiler fuses each into one v_bitop3_b32.)
__device__ __forceinline__ u32 f2key(u32 bits) {
    return bits ^ ((u32)((int)bits >> 31) | 0x80000000u);
}
__device__ __forceinline__ u32 key2f(u32 key) {
    return key ^ (~((u32)((int)key >> 31)) | 0x80000000u);
}

__device__ __forceinline__ void bin_max4(u32* accum, i32x4 s, u32x4 v) {
    atomicMax(&accum[s.x], f2key(v.x));   // -> ds_max_u32 (no-return form)
    atomicMax(&accum[s.y], f2key(v.y));
    atomicMax(&accum[s.z], f2key(v.z));
    atomicMax(&accum[s.w], f2key(v.w));
}

__global__ __launch_bounds__(NTHREADS) void spx_maxpool_kernel(
    const u32* __restrict__ img_bits,   // [B, C, HW] fp32 viewed as bits
    const int* __restrict__ spx,        // [B, HW]
    float* __restrict__ out)            // [B, C, K]
{
    __shared__ u32 accum[K_SZ];         // only LDS object -> offset 0

    const int tid = threadIdx.x;
    const int bc  = blockIdx.x;         // plane id: b*C + c
    const int b   = bc / C_SZ;

    // Init bins to key(-inf).
#pragma unroll
    for (int i = 0; i < K_SZ / NTHREADS; ++i)
        accum[tid + i * NTHREADS] = 0x007FFFFFu;
    __syncthreads();

    const u32x4* __restrict__ img4 =
        (const u32x4*)(img_bits + (size_t)bc * HW_SZ);
    const i32x4* __restrict__ spx4 =
        (const i32x4*)(spx + (size_t)b * HW_SZ);

    // Depth-2 software pipeline, even constant trip count (31) => no
    // remainder guards. Body: clause 4 b128 loads (tiles it, it+1), then
    // 8 ds_max_u32 for tiles it-2, it-1 (compiler can wait loadcnt<=4),
    // then rotate registers.
    u32x4 v0 = __builtin_nontemporal_load(&img4[tid]);
    i32x4 s0 = spx4[tid];
    u32x4 v1 = __builtin_nontemporal_load(&img4[NTHREADS + tid]);
    i32x4 s1 = spx4[NTHREADS + tid];

    for (int it = 2; it < ITERS; it += 2) {
        const int ia = it * NTHREADS + tid;
        const int ib = ia + NTHREADS;
        u32x4 w0 = __builtin_nontemporal_load(&img4[ia]); // read-once stream
        i32x4 t0 = spx4[ia];                              // L2-hot labels
        u32x4 w1 = __builtin_nontemporal_load(&img4[ib]);
        i32x4 t1 = spx4[ib];
        bin_max4(accum, s0, v0);
        bin_max4(accum, s1, v1);
        v0 = w0; s0 = t0;
        v1 = w1; s1 = t1;
    }
    bin_max4(accum, s0, v0);
    bin_max4(accum, s1, v1);
    __syncthreads();

    // Unmap keys -> float bits in place (each thread owns its own 16B slice).
    u32x4* acc4 = (u32x4*)accum;
    u32x4 kv = acc4[tid];
    kv.x = key2f(kv.x);
    kv.y = key2f(kv.y);
    kv.z = key2f(kv.z);
    kv.w = key2f(kv.w);
    acc4[tid] = kv;

    // Drain LDS -> global via the CDNA5 async data mover (16B per lane).
    // s_wait_dscnt 0: our ds_store above must land before the async engine
    // reads LDS (async ops are unordered vs DS). s_wait_asynccnt 0 before
    // wave exit (s_endpgm also waits-idle, but be explicit).
    unsigned long long gdst =
        (unsigned long long)(uintptr_t)(out + (size_t)bc * K_SZ + tid * 4);
    u32 lds_off = (u32)(tid * 16);      // accum[] sits at LDS offset 0
    asm volatile(
        "s_wait_dscnt 0x0\n\t"
        "global_store_async_from_lds_b128 %0, %1, off\n\t"
        "s_wait_asynccnt 0x0"
        :
        : "v"(gdst), "v"(lds_off)
        : "memory");
}

extern "C" void kernel_launch(void* const* d_in, const int* in_sizes, int n_in,
                              void* d_out, int out_size, void* d_ws, size_t ws_size,
                              hipStream_t stream) {
    (void)in_sizes; (void)n_in; (void)d_ws; (void)ws_size; (void)out_size;
    const u32* img = (const u32*)d_in[0];   // fp32 bits of img [B,C,H,W]
    const int* spx = (const int*)d_in[1];   // int32 labels  [B,H,W]
    float* out = (float*)d_out;             // fp32 out      [B,C,K]

    dim3 grid(B_SZ * C_SZ);                 // 1024 planes, one WG each
    dim3 block(NTHREADS);
    hipLaunchKernelGGL(spx_maxpool_kernel, grid, block, 0, stream,
                       img, spx, out);
}